// S3GAT_28157805593446
// MI455X (gfx1250) — compile-verified
//
#include <hip/hip_runtime.h>
#include <hip/hip_bf16.h>
#include <math.h>

// ---------------------------------------------------------------------------
// S3GAT on gfx1250: 3-layer GAT, fp32 WMMA for the dense projections,
// atomic segment-softmax for the edge phase.
// ---------------------------------------------------------------------------

typedef float v2f __attribute__((ext_vector_type(2)));
typedef float v8f __attribute__((ext_vector_type(8)));

#define GAT_N   50000
#define GAT_IN  128
#define GAT_HD  128   // heads*dim for layers 0/1
#define GAT_OUT 40

// Monotonic float -> uint key so unsigned atomicMax == float max.
__device__ __forceinline__ unsigned orderKey(float f) {
    unsigned u = __float_as_uint(f);
    return (u & 0x80000000u) ? ~u : (u | 0x80000000u);
}
__device__ __forceinline__ float keyToFloat(unsigned k) {
    unsigned u = (k & 0x80000000u) ? (k ^ 0x80000000u) : ~k;
    return __uint_as_float(u);
}
#define MIN_KEY 0x007FFFFFu  // orderKey(-inf)

// ---------------------------------------------------------------------------
// C[n,kout] = A[n,kin] * W[kin,kout], fp32 WMMA 16x16x4.
// One wave32 computes a 16 x (16*NT) output tile; A fragment reused NT times.
// Rows: launch guarantees n % 16 == 0 (N = 50000 = 3125*16).
// GUARD=true adds column bounds checks (layer 2, kout=40).
// A layout: lane<16 -> row=lane, K={kk,kk+1}; lane>=16 -> row=lane-16, K={kk+2,kk+3}
// B layout: VGPR0 = K rows {kb}, VGPR1 = {kb+1}; C: VGPR j = row j / j+8, col=lane&15
// ---------------------------------------------------------------------------
template <int NT, bool GUARD>
__global__ void gat_gemm_wmma(const float* __restrict__ A,
                              const float* __restrict__ W,
                              float* __restrict__ C,
                              int n, int kin, int kout, int colGroups) {
    const int wid  = blockIdx.x * (blockDim.x >> 5) + (threadIdx.x >> 5);
    const int lane = threadIdx.x & 31;
    const int tileM = wid / colGroups;
    const int tileN = wid % colGroups;
    const int rowBase = tileM * 16;
    if (rowBase >= n) return;               // wave-uniform exit
    const int colBase = tileN * (16 * NT);

    const int r    = lane & 15;
    const int kb   = (lane >> 4) << 1;      // 0 or 2
    const int col  = lane & 15;
    const int grow = rowBase + r;

    v8f acc[NT];
    #pragma unroll
    for (int t = 0; t < NT; ++t) acc[t] = (v8f){};

    for (int kk = 0; kk < kin; kk += 4) {
        v2f a;
        a.x = A[(size_t)grow * kin + kk + kb];
        a.y = A[(size_t)grow * kin + kk + kb + 1];
        #pragma unroll
        for (int t = 0; t < NT; ++t) {
            const int gc = colBase + 16 * t + col;
            v2f b;
            if (GUARD) {
                const bool ok = gc < kout;
                const int  bc = ok ? gc : 0;
                float b0 = W[(size_t)(kk + kb)     * kout + bc];
                float b1 = W[(size_t)(kk + kb + 1) * kout + bc];
                b.x = ok ? b0 : 0.0f;
                b.y = ok ? b1 : 0.0f;
            } else {
                b.x = W[(size_t)(kk + kb)     * kout + gc];
                b.y = W[(size_t)(kk + kb + 1) * kout + gc];
            }
            acc[t] = __builtin_amdgcn_wmma_f32_16x16x4_f32(
                /*neg_a=*/false, a, /*neg_b=*/false, b,
                /*c_mod=*/(short)0, acc[t], /*reuse_a=*/false, /*reuse_b=*/false);
        }
    }

    const int mbase = (lane >> 4) << 3;     // 0 or 8
    #pragma unroll
    for (int t = 0; t < NT; ++t) {
        const int gc = colBase + 16 * t + col;
        if (!GUARD || gc < kout) {
            #pragma unroll
            for (int j = 0; j < 8; ++j) {
                int gm = rowBase + mbase + j;
                if (gm < n) C[(size_t)gm * kout + gc] = acc[t][j];
            }
        }
    }
}

// al_s[n,h] = sum_d h[n,h,d]*as[h,d] ; al_d likewise.
__global__ void gat_attn_logits(const float* __restrict__ h,
                                const float* __restrict__ as_,
                                const float* __restrict__ ad_,
                                float* __restrict__ alS,
                                float* __restrict__ alD,
                                int n, int heads, int d) {
    int t = blockIdx.x * blockDim.x + threadIdx.x;
    if (t >= n * heads) return;
    int node = t / heads, hh = t % heads;
    const float* hp  = h + ((size_t)node * heads + hh) * d;
    const float* asp = as_ + (size_t)hh * d;
    const float* adp = ad_ + (size_t)hh * d;
    float ss = 0.f, dd = 0.f;
    for (int i = 0; i < d; ++i) { float v = hp[i]; ss += v * asp[i]; dd += v * adp[i]; }
    alS[t] = ss; alD[t] = dd;
}

// zero/seed all per-layer accumulators
__global__ void gat_init(unsigned* __restrict__ mKey, float* __restrict__ sBuf,
                         float* __restrict__ agg, float* __restrict__ bnSum,
                         float* __restrict__ bnSqs, int nH, int nHD) {
    int t = blockIdx.x * blockDim.x + threadIdx.x;
    if (t < nH)  { mKey[t] = MIN_KEY; sBuf[t] = 0.f; }
    if (t < nHD) { agg[t] = 0.f; }
    if (bnSum != nullptr && t < GAT_HD) { bnSum[t] = 0.f; bnSqs[t] = 0.f; }
}

// pass 1: e = leaky_relu(alS[src]+alD[dst]); store; segment max via key atomicMax
__global__ void gat_edge_max(const int* __restrict__ src, const int* __restrict__ dst,
                             const float* __restrict__ alS, const float* __restrict__ alD,
                             float* __restrict__ eBuf, unsigned* __restrict__ mKey,
                             int E, int heads) {
    int t = blockIdx.x * blockDim.x + threadIdx.x;
    if (t >= E * heads) return;
    int e = t / heads, hh = t % heads;
    int s = src[e], dd = dst[e];
    float v = alS[(size_t)s * heads + hh] + alD[(size_t)dd * heads + hh];
    v = (v > 0.f) ? v : 0.2f * v;           // leaky_relu slope 0.2
    eBuf[t] = v;
    atomicMax(&mKey[(size_t)dd * heads + hh], orderKey(v));
}

// pass 2: p = exp(e - m[dst]); segment sum
__global__ void gat_edge_expsum(const int* __restrict__ dst,
                                const float* __restrict__ eBuf,
                                const unsigned* __restrict__ mKey,
                                float* __restrict__ sBuf, int E, int heads) {
    int t = blockIdx.x * blockDim.x + threadIdx.x;
    if (t >= E * heads) return;
    int e = t / heads, hh = t % heads;
    int dd = dst[e];
    float m = keyToFloat(mKey[(size_t)dd * heads + hh]);
    if (!__builtin_isfinite(m)) m = 0.f;
    float p = expf(eBuf[t] - m);
    atomicAdd(&sBuf[(size_t)dd * heads + hh], p);
}

// pass 2b: convert logits to final alpha in place (removes transcendentals from
// the bandwidth-bound aggregation kernel)
__global__ void gat_edge_alpha(const int* __restrict__ dst,
                               float* __restrict__ eBuf,
                               const unsigned* __restrict__ mKey,
                               const float* __restrict__ sBuf, int E, int heads) {
    int t = blockIdx.x * blockDim.x + threadIdx.x;
    if (t >= E * heads) return;
    int e = t / heads, hh = t % heads;
    int dd = dst[e];
    float m = keyToFloat(mKey[(size_t)dd * heads + hh]);
    if (!__builtin_isfinite(m)) m = 0.f;
    float denom = sBuf[(size_t)dd * heads + hh] + 1e-16f;
    eBuf[t] = expf(eBuf[t] - m) / denom;
}

// pass 3: wave per edge; agg[dst,:] += alpha * h[src,:]  (pure gather+atomic)
__global__ void gat_edge_aggregate(const int* __restrict__ src, const int* __restrict__ dst,
                                   const float* __restrict__ h,
                                   const float* __restrict__ alpha,
                                   float* __restrict__ agg,
                                   int E, int heads, int d, int hd) {
    int wave = blockIdx.x * (blockDim.x >> 5) + (threadIdx.x >> 5);
    int lane = threadIdx.x & 31;
    if (wave >= E) return;
    int s = src[wave], dd = dst[wave];
    for (int c = lane; c < hd; c += 32) {
        int hh = c / d;
        float a = alpha[(size_t)wave * heads + hh];
        atomicAdd(&agg[(size_t)dd * hd + c], a * h[(size_t)s * hd + c]);
    }
}

// y = agg + bias; per-channel partial sums for batchnorm (blockDim == 128 channels)
__global__ void gat_bias_bnreduce(const float* __restrict__ agg,
                                  const float* __restrict__ bias,
                                  float* __restrict__ y,
                                  float* __restrict__ bnSum, float* __restrict__ bnSqs,
                                  int n, int rows) {
    int c  = threadIdx.x;                    // 0..127
    int r0 = blockIdx.x * rows;
    int r1 = min(r0 + rows, n);
    float b = bias[c], s = 0.f, s2 = 0.f;
    for (int r = r0; r < r1; ++r) {
        float v = agg[(size_t)r * GAT_HD + c] + b;
        y[(size_t)r * GAT_HD + c] = v;
        s += v; s2 += v * v;
    }
    atomicAdd(&bnSum[c], s);
    atomicAdd(&bnSqs[c], s2);
}

// act = elu(batchnorm(y)) * beta_scale
__global__ void gat_bn_elu(const float* __restrict__ y,
                           const float* __restrict__ g, const float* __restrict__ be,
                           const float* __restrict__ bnSum, const float* __restrict__ bnSqs,
                           float* __restrict__ act, int n, float scale) {
    int t = blockIdx.x * blockDim.x + threadIdx.x;
    if (t >= n * GAT_HD) return;
    int c = t & (GAT_HD - 1);
    const float invN = 1.0f / (float)GAT_N;
    float mu  = bnSum[c] * invN;
    float var = bnSqs[c] * invN - mu * mu;
    float v = g[c] * (y[t] - mu) * rsqrtf(var + 1e-5f) + be[c];
    v = (v > 0.f) ? v : (expf(v) - 1.f);     // elu
    act[t] = v * scale;
}

// out = log_softmax(agg + bias) over 40 classes
__global__ void gat_final_logsoftmax(const float* __restrict__ agg,
                                     const float* __restrict__ bias,
                                     float* __restrict__ out, int n) {
    int node = blockIdx.x * blockDim.x + threadIdx.x;
    if (node >= n) return;
    const float* a = agg + (size_t)node * GAT_OUT;
    float mx = -INFINITY;
    for (int c = 0; c < GAT_OUT; ++c) { float v = a[c] + bias[c]; mx = fmaxf(mx, v); }
    float s = 0.f;
    for (int c = 0; c < GAT_OUT; ++c) s += expf(a[c] + bias[c] - mx);
    float lse = mx + logf(s);
    for (int c = 0; c < GAT_OUT; ++c) out[(size_t)node * GAT_OUT + c] = a[c] + bias[c] - lse;
}

// ---------------------------------------------------------------------------
extern "C" void kernel_launch(void* const* d_in, const int* in_sizes, int n_in,
                              void* d_out, int out_size, void* d_ws, size_t ws_size,
                              hipStream_t stream) {
    const float* x   = (const float*)d_in[0];
    const int*   ei  = (const int*)d_in[1];
    const int N = in_sizes[0] / GAT_IN;
    const int E = in_sizes[1] / 2;
    const int* srcIdx = ei;
    const int* dstIdx = ei + E;

    const float* W[3]  = { (const float*)d_in[2],  (const float*)d_in[6],  (const float*)d_in[10] };
    const float* AS[3] = { (const float*)d_in[3],  (const float*)d_in[7],  (const float*)d_in[11] };
    const float* AD[3] = { (const float*)d_in[4],  (const float*)d_in[8],  (const float*)d_in[12] };
    const float* B[3]  = { (const float*)d_in[5],  (const float*)d_in[9],  (const float*)d_in[13] };
    const float* GBN[2] = { (const float*)d_in[14], (const float*)d_in[16] };
    const float* BBN[2] = { (const float*)d_in[15], (const float*)d_in[17] };

    // workspace layout (fp32 elements)
    float* f = (float*)d_ws;
    float*    bufH   = f;                         // N*128 (reused as Y post-bias)
    float*    bufAct = bufH   + (size_t)N * GAT_HD;
    float*    bufAgg = bufAct + (size_t)N * GAT_HD;
    float*    eBuf   = bufAgg + (size_t)N * GAT_HD; // E*4 max (logits, then alpha)
    float*    alS    = eBuf   + (size_t)E * 4;
    float*    alD    = alS    + (size_t)N * 4;
    float*    sBuf   = alD    + (size_t)N * 4;
    float*    bnSum  = sBuf   + (size_t)N * 4;
    float*    bnSqs  = bnSum  + GAT_HD;
    unsigned* mKey   = (unsigned*)(bnSqs + GAT_HD);  // N*4

    const int heads[3] = { 4, 4, 1 };
    const int dim[3]   = { 32, 32, GAT_OUT };
    const int kout[3]  = { GAT_HD, GAT_HD, GAT_OUT };
    const float betaScale[2] = { 1.0f + 1.0f * (1.0f / (25.0f + 1e-3f)),
                                 1.0f + 0.5f * (1.0f / (25.0f + 1e-3f)) };

    const float* layerIn = x;
    for (int l = 0; l < 3; ++l) {
        const int H = heads[l], D = dim[l], HD = kout[l];

        // 1) projection GEMM (fp32 WMMA)
        if (l < 2) {
            // kout=128: unguarded, 16x64 tile per wave (NT=4)
            int colGroups = HD / 64;                         // = 2
            int tiles = (N / 16) * colGroups;
            int blocks = (tiles + 7) / 8;                    // 8 waves / block
            gat_gemm_wmma<4, false><<<blocks, 256, 0, stream>>>(
                layerIn, W[l], bufH, N, GAT_IN, HD, colGroups);
        } else {
            // kout=40: guarded, 16x16 tile per wave
            int colGroups = (HD + 15) / 16;                  // = 3
            int tiles = ((N + 15) / 16) * colGroups;
            int blocks = (tiles + 7) / 8;
            gat_gemm_wmma<1, true><<<blocks, 256, 0, stream>>>(
                layerIn, W[l], bufH, N, GAT_IN, HD, colGroups);
        }
        // 2) per-node attention logits
        {
            int th = N * H;
            gat_attn_logits<<<(th + 255) / 256, 256, 0, stream>>>(bufH, AS[l], AD[l],
                                                                  alS, alD, N, H, D);
        }
        // 3) init accumulators
        {
            int th = N * HD;
            float* bnS = (l < 2) ? bnSum : nullptr;
            float* bnQ = (l < 2) ? bnSqs : nullptr;
            gat_init<<<(th + 255) / 256, 256, 0, stream>>>(mKey, sBuf, bufAgg,
                                                           bnS, bnQ, N * H, th);
        }
        // 4) segment max / 5) exp-sum / 5b) alpha / 6) aggregate
        {
            int th = E * H;
            gat_edge_max<<<(th + 255) / 256, 256, 0, stream>>>(srcIdx, dstIdx, alS, alD,
                                                               eBuf, mKey, E, H);
            gat_edge_expsum<<<(th + 255) / 256, 256, 0, stream>>>(dstIdx, eBuf, mKey,
                                                                  sBuf, E, H);
            gat_edge_alpha<<<(th + 255) / 256, 256, 0, stream>>>(dstIdx, eBuf, mKey,
                                                                 sBuf, E, H);
            int blocks = (E + 7) / 8;               // wave per edge
            gat_edge_aggregate<<<blocks, 256, 0, stream>>>(srcIdx, dstIdx, bufH, eBuf,
                                                           bufAgg, E, H, D, HD);
        }
        if (l < 2) {
            // 7) bias + BN partial sums (bufH reused as pre-BN activations)
            int rows = 64;
            gat_bias_bnreduce<<<(N + rows - 1) / rows, GAT_HD, 0, stream>>>(
                bufAgg, B[l], bufH, bnSum, bnSqs, N, rows);
            // 8) BN + ELU + beta scale
            int th = N * GAT_HD;
            gat_bn_elu<<<(th + 255) / 256, 256, 0, stream>>>(bufH, GBN[l], BBN[l],
                                                             bnSum, bnSqs, bufAct,
                                                             N, betaScale[l]);
            layerIn = bufAct;
        } else {
            // 9) bias + log_softmax -> d_out
            gat_final_logsoftmax<<<(N + 255) / 256, 256, 0, stream>>>(
                bufAgg, B[l], (float*)d_out, N);
        }
    }
}